// GCNRegressor_4243427688732
// MI455X (gfx1250) — compile-verified
//
#include <hip/hip_runtime.h>

typedef __attribute__((ext_vector_type(2))) float v2f;
typedef __attribute__((ext_vector_type(8))) float v8f;

#define HID 32

// ---------------- utility ----------------
__global__ void fill_f32(float* __restrict__ p, float v, long n) {
  long i = (long)blockIdx.x * blockDim.x + threadIdx.x;
  long stride = (long)gridDim.x * blockDim.x;
  for (; i < n; i += stride) p[i] = v;
}

// deg[col[e]] += w[e]   (deg pre-filled with 1.0 = self-loop weight)
__global__ void deg_accum(const int* __restrict__ col, const float* __restrict__ w,
                          float* __restrict__ deg, long E) {
  long e = (long)blockIdx.x * blockDim.x + threadIdx.x;
  if (e < E) atomicAdd(&deg[col[e]], w[e]);
}

// deg -> dinv in place
__global__ void dinv_kernel(float* __restrict__ deg, long n) {
  long i = (long)blockIdx.x * blockDim.x + threadIdx.x;
  if (i < n) {
    float d = deg[i];
    deg[i] = d > 0.0f ? rsqrtf(d) : 0.0f;
  }
}

// layer-0 GEMM: h[i,f] = sum_k x[i,k] * W0[k,f]   (K=3, scalar; lane = feature)
__global__ void gemm0(const float* __restrict__ x, const float* __restrict__ W0,
                      float* __restrict__ h, long N) {
  long gid = (long)blockIdx.x * blockDim.x + threadIdx.x;
  long i = gid >> 5;
  int  f = (int)(gid & 31);
  if (i < N) {
    float v = x[i*3+0]*W0[f] + x[i*3+1]*W0[HID+f] + x[i*3+2]*W0[2*HID+f];
    h[i*HID + f] = v;
  }
}

// per-edge scatter: one wave per edge, lane = feature
__global__ void scatter(const int* __restrict__ row, const int* __restrict__ col,
                        const float* __restrict__ w, const float* __restrict__ dinv,
                        const float* __restrict__ h, float* __restrict__ agg, long E) {
  long gid = (long)blockIdx.x * blockDim.x + threadIdx.x;
  long e = gid >> 5;
  int  f = (int)(gid & 31);
  if (e < E) {
    int r = row[e];
    int c = col[e];
    float nrm = dinv[r] * w[e] * dinv[c];
    atomicAdd(&agg[(long)c * HID + f], h[(long)r * HID + f] * nrm);
  }
}

// out = relu(agg + dinv^2 * h_selfloop + b)
__global__ void bias_relu(const float* __restrict__ agg, const float* __restrict__ h,
                          const float* __restrict__ dinv, const float* __restrict__ b,
                          float* __restrict__ out, long N) {
  long gid = (long)blockIdx.x * blockDim.x + threadIdx.x;
  long i = gid >> 5;
  int  f = (int)(gid & 31);
  if (i < N) {
    float di = dinv[i];
    float v = agg[i*HID + f] + di * di * h[i*HID + f] + b[f];
    out[i*HID + f] = v > 0.0f ? v : 0.0f;
  }
}

// layer-1 GEMM via V_WMMA_F32_16X16X4_F32: out[Nx32] = h[Nx32] @ W[32x32]
// One wave per 16x16 output tile; K=32 accumulated in 8 chunks of 4.
__global__ void gemm1_wmma(const float* __restrict__ h, const float* __restrict__ W,
                           float* __restrict__ out, int numRowTiles) {
  int wave = (int)((blockIdx.x * (long)blockDim.x + threadIdx.x) >> 5);
  int lane = threadIdx.x & 31;
  int totalTiles = numRowTiles * 2;           // 2 N-tiles of 16
  if (wave >= totalTiles) return;             // wave-uniform guard (EXEC all-1s inside)
  int mt = wave >> 1, nt = wave & 1;
  long m0 = (long)mt * 16;
  int  n0 = nt * 16;
  int  half = lane >> 4;                      // 0: lanes 0-15, 1: lanes 16-31
  int  l    = lane & 15;

  v8f c = {};
#pragma unroll
  for (int kk = 0; kk < HID; kk += 4) {
    // A fragment (16x4 f32): lanes 0-15 carry K = kk,kk+1 ; lanes 16-31 K = kk+2,kk+3
    int ka = kk + (half ? 2 : 0);
    v2f a, b;
    a.x = h[(m0 + l) * HID + ka];
    a.y = h[(m0 + l) * HID + ka + 1];
    // B fragment (4x16 f32): same K split across lane halves, N = n0 + l
    b.x = W[ka * HID + n0 + l];
    b.y = W[(ka + 1) * HID + n0 + l];
    c = __builtin_amdgcn_wmma_f32_16x16x4_f32(false, a, false, b, (short)0, c,
                                              false, false);
  }
  // D layout: VGPR r -> row m0 + r + 8*half, col n0 + l
#pragma unroll
  for (int r = 0; r < 8; ++r)
    out[(m0 + r + half * 8) * HID + n0 + l] = c[r];
}

// scalar remainder for rows not covered by whole 16-row tiles (N%16 != 0)
__global__ void gemm1_tail(const float* __restrict__ h, const float* __restrict__ W,
                           float* __restrict__ out, long rowStart, long N) {
  long gid = (long)blockIdx.x * blockDim.x + threadIdx.x;
  long i = rowStart + (gid >> 5);
  int  f = (int)(gid & 31);
  if (i < N) {
    float s = 0.0f;
    for (int k = 0; k < HID; ++k) s += h[i*HID + k] * W[k*HID + f];
    out[i*HID + f] = s;
  }
}

// mean-pool accumulation: wave per node, lane = feature
__global__ void pool(const float* __restrict__ h2, const int* __restrict__ batch,
                     float* __restrict__ pooled, float* __restrict__ cnt, long N) {
  long gid = (long)blockIdx.x * blockDim.x + threadIdx.x;
  long i = gid >> 5;
  int  f = (int)(gid & 31);
  if (i < N) {
    int g = batch[i];
    atomicAdd(&pooled[(long)g * HID + f], h2[i*HID + f]);
    if (f == 0) atomicAdd(&cnt[g], 1.0f);
  }
}

// out[g] = dot(pooled[g], Wr) / max(cnt[g],1) + br
__global__ void readout(const float* __restrict__ pooled, const float* __restrict__ cnt,
                        const float* __restrict__ Wr, const float* __restrict__ br,
                        float* __restrict__ out, int G) {
  int g = blockIdx.x * blockDim.x + threadIdx.x;
  if (g < G) {
    float s = 0.0f;
#pragma unroll
    for (int f = 0; f < HID; ++f) s += pooled[(long)g * HID + f] * Wr[f];
    float cc = cnt[g];
    cc = cc > 1.0f ? cc : 1.0f;
    out[g] = s / cc + br[0];
  }
}

static inline size_t alignup(size_t x) { return (x + 255) & ~(size_t)255; }

extern "C" void kernel_launch(void* const* d_in, const int* in_sizes, int n_in,
                              void* d_out, int out_size, void* d_ws, size_t ws_size,
                              hipStream_t stream) {
  const float* x     = (const float*)d_in[0];
  const int*   ei    = (const int*)  d_in[1];
  const float* ew    = (const float*)d_in[2];
  const int*   batch = (const int*)  d_in[3];
  const float* W0    = (const float*)d_in[4];
  const float* b0    = (const float*)d_in[5];
  const float* W1    = (const float*)d_in[6];
  const float* b1    = (const float*)d_in[7];
  const float* Wr    = (const float*)d_in[8];
  const float* br    = (const float*)d_in[9];
  float* out = (float*)d_out;

  long N = in_sizes[0] / 3;
  long E = in_sizes[2];
  int  G = out_size;
  const int* row = ei;        // edge_index[0]
  const int* col = ei + E;    // edge_index[1]

  char* ws = (char*)d_ws;
  float* bufH   = (float*)ws; ws += alignup((size_t)N * HID * sizeof(float));
  float* bufAgg = (float*)ws; ws += alignup((size_t)N * HID * sizeof(float));
  float* bufH2  = (float*)ws; ws += alignup((size_t)N * HID * sizeof(float));
  float* dinv   = (float*)ws; ws += alignup((size_t)N * sizeof(float));
  float* pooled = (float*)ws; ws += alignup((size_t)G * HID * sizeof(float));
  float* cnt    = (float*)ws;

  dim3 blk(256);
  auto cdiv  = [](long a, long b) { return (a + b - 1) / b; };
  auto gridf = [&](long n) { long g = cdiv(n, 256); return (unsigned)(g > 8192 ? 8192 : g); };

  // degree / dinv (deg starts at 1.0 = self-loop weight)
  fill_f32<<<gridf(N), blk, 0, stream>>>(dinv, 1.0f, N);
  deg_accum<<<(unsigned)cdiv(E, 256), blk, 0, stream>>>(col, ew, dinv, E);
  dinv_kernel<<<(unsigned)cdiv(N, 256), blk, 0, stream>>>(dinv, N);

  // ---- layer 0 ----
  gemm0<<<(unsigned)cdiv(N * 32, 256), blk, 0, stream>>>(x, W0, bufH, N);
  fill_f32<<<gridf(N * HID), blk, 0, stream>>>(bufAgg, 0.0f, N * HID);
  scatter<<<(unsigned)cdiv(E * 32, 256), blk, 0, stream>>>(row, col, ew, dinv, bufH, bufAgg, E);
  bias_relu<<<(unsigned)cdiv(N * 32, 256), blk, 0, stream>>>(bufAgg, bufH, dinv, b0, bufH2, N);

  // ---- layer 1: WMMA fp32 GEMM bufH2 @ W1 -> bufH ----
  long numRowTiles = N / 16;
  if (numRowTiles > 0) {
    long totalTiles = numRowTiles * 2;            // 8 waves (tiles) per 256-thread block
    gemm1_wmma<<<(unsigned)cdiv(totalTiles, 8), blk, 0, stream>>>(bufH2, W1, bufH,
                                                                  (int)numRowTiles);
  }
  long tailRows = N - numRowTiles * 16;
  if (tailRows > 0)
    gemm1_tail<<<(unsigned)cdiv(tailRows * 32, 256), blk, 0, stream>>>(bufH2, W1, bufH,
                                                                       numRowTiles * 16, N);
  fill_f32<<<gridf(N * HID), blk, 0, stream>>>(bufAgg, 0.0f, N * HID);
  scatter<<<(unsigned)cdiv(E * 32, 256), blk, 0, stream>>>(row, col, ew, dinv, bufH, bufAgg, E);
  bias_relu<<<(unsigned)cdiv(N * 32, 256), blk, 0, stream>>>(bufAgg, bufH, dinv, b1, bufH2, N);

  // ---- pooling + readout ----
  fill_f32<<<gridf(G * HID), blk, 0, stream>>>(pooled, 0.0f, (long)G * HID);
  fill_f32<<<gridf(G), blk, 0, stream>>>(cnt, 0.0f, G);
  pool<<<(unsigned)cdiv(N * 32, 256), blk, 0, stream>>>(bufH2, batch, pooled, cnt, N);
  readout<<<(unsigned)cdiv(G, 256), blk, 0, stream>>>(pooled, cnt, Wr, br, out, G);
}